// controller_78632261255249
// MI455X (gfx1250) — compile-verified
//
#include <hip/hip_runtime.h>

#define HID 10
#define GHID 30           // 3*HID gate rows
#define NOUT 3
#define MAXN 10000

typedef float v2f __attribute__((ext_vector_type(2)));
typedef float v8f __attribute__((ext_vector_type(8)));

__device__ __forceinline__ float rdlane(float v, int l) {
  return __int_as_float(__builtin_amdgcn_readlane(__float_as_int(v), l));
}
__device__ __forceinline__ float fexp(float x) {
  return __builtin_amdgcn_exp2f(x * 1.4426950408889634f);   // v_exp_f32 computes 2^x
}
__device__ __forceinline__ float fsig(float x) {
  return __builtin_amdgcn_rcpf(1.0f + fexp(-x));            // sigmoid
}
__device__ __forceinline__ float ftanh(float x) {
  return fmaf(2.0f, __builtin_amdgcn_rcpf(1.0f + fexp(-2.0f * x)), -1.0f);
}

__global__ __launch_bounds__(32)
void controller_gru_kernel(
    const float* __restrict__ state,
    const float* __restrict__ Wihe, const float* __restrict__ Whhe,
    const float* __restrict__ bihe, const float* __restrict__ bhhe,
    const float* __restrict__ Wihd, const float* __restrict__ Whhd,
    const float* __restrict__ bihd, const float* __restrict__ bhhd,
    const float* __restrict__ Wout, const float* __restrict__ bout,
    float* __restrict__ out, int n)
{
  __shared__ float s_state[MAXN];
  __shared__ float s_cvec[32];

  const int lane = threadIdx.x;
  if (n > MAXN) n = MAXN;
  if (n <= 0) return;

  // ---- stage the state stream into LDS (one pass, full-wave b128 copies) ----
  {
    const int nv4 = n >> 2;
    const float4* g4 = (const float4*)state;
    float4* l4 = (float4*)s_state;
    for (int i = lane; i < nv4; i += 32) l4[i] = g4[i];
    for (int i = (nv4 << 2) + lane; i < n; i += 32) s_state[i] = state[i];
  }

  const int ii = (lane < HID) ? lane : 0;   // clamp: lanes >= 10 shadow lane 0

  // ---- encoder weights into VGPRs (lane ii owns gate rows ii, 10+ii, 20+ii) ----
  float wr[HID], wz[HID], wn[HID];
#pragma unroll
  for (int k = 0; k < HID; ++k) {
    wr[k] = Whhe[ii * HID + k];
    wz[k] = Whhe[(HID + ii) * HID + k];
    wn[k] = Whhe[(2 * HID + ii) * HID + k];
  }
  const float ur = Wihe[ii], uz = Wihe[HID + ii], un = Wihe[2 * HID + ii];
  const float crc = bihe[ii] + bhhe[ii];
  const float czc = bihe[HID + ii] + bhhe[HID + ii];
  const float cin = bihe[2 * HID + ii];
  const float chn = bhhe[2 * HID + ii];

  __syncthreads();

  // ---- encoder GRU scan (sequential; lanes 0-9 active, h broadcast by readlane) ----
  float hown = 0.0f;       // this lane's h[ii]
  float hb[HID];           // h broadcast to all lanes (SGPR-resident via readlane)
#pragma unroll
  for (int k = 0; k < HID; ++k) hb[k] = 0.0f;

  float xc = s_state[0];
  for (int t = 0; t < n; ++t) {
    const float xn = s_state[(t + 1 < n) ? (t + 1) : t];  // software prefetch
    float aR = fmaf(ur, xc, crc);
    float aZ = fmaf(uz, xc, czc);
    float aN = chn;
#pragma unroll
    for (int k = 0; k < HID; ++k) {     // three independent FMA chains
      aR = fmaf(wr[k], hb[k], aR);
      aZ = fmaf(wz[k], hb[k], aZ);
      aN = fmaf(wn[k], hb[k], aN);
    }
    const float r = fsig(aR);
    const float z = fsig(aZ);
    const float nn = ftanh(fmaf(r, aN, fmaf(un, xc, cin)));
    hown = fmaf(z, hown - nn, nn);      // (1-z)*n + z*h
#pragma unroll
    for (int k = 0; k < HID; ++k) hb[k] = rdlane(hown, k);
    xc = xn;
  }
  // hb[0..9] == c (encoder final hidden); hown == c[ii] per lane.

  // ---- one-time cvec = W_ih_d[:,1:11] @ c via V_WMMA_F32_16X16X4_F32 ----
  // Orientation: D = A(16x4, row0 = c chunk) x B(4x16, W^T chunk). D row 0 = cvec tile.
  {
    const int half = lane >> 4;        // 0: lanes 0-15, 1: lanes 16-31
    const int col  = lane & 15;
    v8f acc0 = {};                     // cvec[0..15]
    v8f acc1 = {};                     // cvec[16..31]
#pragma unroll
    for (int k0 = 0; k0 < 12; k0 += 4) {
      // A 16x4 f32 layout: lanes 0-15 hold K=0(v0),K=1(v1); lanes 16-31 hold K=2,K=3.
      const float c0 = (k0 + 0 < HID) ? hb[k0 + 0] : 0.0f;
      const float c1 = (k0 + 1 < HID) ? hb[(k0 + 1 < HID) ? k0 + 1 : 0] : 0.0f;
      const float c2 = (k0 + 2 < HID) ? hb[(k0 + 2 < HID) ? k0 + 2 : 0] : 0.0f;
      const float c3 = (k0 + 3 < HID) ? hb[(k0 + 3 < HID) ? k0 + 3 : 0] : 0.0f;
      v2f A;
      A.x = (lane == 0) ? c0 : ((lane == 16) ? c2 : 0.0f);
      A.y = (lane == 0) ? c1 : ((lane == 16) ? c3 : 0.0f);
      // B 4x16 f32: v0 = rows {k0, k0+2}, v1 = rows {k0+1, k0+3}, col = lane&15.
      const int kv0 = k0 + (half ? 2 : 0);
      const int kv1 = k0 + (half ? 3 : 1);
      const int j0 = col, j1 = 16 + col;
      v2f B0, B1;
      B0.x = (kv0 < HID) ? Wihd[j0 * (1 + HID) + 1 + kv0] : 0.0f;
      B0.y = (kv1 < HID) ? Wihd[j0 * (1 + HID) + 1 + kv1] : 0.0f;
      B1.x = (j1 < GHID && kv0 < HID) ? Wihd[j1 * (1 + HID) + 1 + kv0] : 0.0f;
      B1.y = (j1 < GHID && kv1 < HID) ? Wihd[j1 * (1 + HID) + 1 + kv1] : 0.0f;
      acc0 = __builtin_amdgcn_wmma_f32_16x16x4_f32(false, A, false, B0,
                                                   (short)0, acc0, false, false);
      acc1 = __builtin_amdgcn_wmma_f32_16x16x4_f32(false, A, false, B1,
                                                   (short)0, acc1, false, false);
    }
    // D f32 16x16 layout: VGPR0 lanes 0-15 = row M=0 (our result vector).
    if (lane < 16) {
      s_cvec[lane]      = acc0[0];
      s_cvec[16 + lane] = acc1[0];
    }
  }
  __syncthreads();

  // ---- decoder weights; fold biases + cvec (attention is identity: ctx == c) ----
  float vr[HID], vz[HID], vn[HID], wo[HID];
#pragma unroll
  for (int k = 0; k < HID; ++k) {
    vr[k] = Whhd[ii * HID + k];
    vz[k] = Whhd[(HID + ii) * HID + k];
    vn[k] = Whhd[(2 * HID + ii) * HID + k];
  }
  const float sr = Wihd[ii * (1 + HID)];
  const float sz = Wihd[(HID + ii) * (1 + HID)];
  const float sn = Wihd[(2 * HID + ii) * (1 + HID)];
  const float erc = bihd[ii] + bhhd[ii] + s_cvec[ii];
  const float ezc = bihd[HID + ii] + bhhd[HID + ii] + s_cvec[HID + ii];
  const float einc = bihd[2 * HID + ii] + s_cvec[2 * HID + ii];
  const float ehn = bhhd[2 * HID + ii];

  const int oo = (lane < NOUT) ? lane : 0;
#pragma unroll
  for (int k = 0; k < HID; ++k) wo[k] = Wout[oo * HID + k];
  const float bo = bout[oo];

  if (lane < NOUT) out[lane] = 0.0f;   // logits[0] stays zero

  // ---- decoder GRU scan, t = 1 .. n-1 (hidden init = c) ----
  float xc2 = (n > 1) ? s_state[1] : 0.0f;
  for (int t = 1; t < n; ++t) {
    const float xn = s_state[(t + 1 < n) ? (t + 1) : t];
    float aR = fmaf(sr, xc2, erc);
    float aZ = fmaf(sz, xc2, ezc);
    float aN = ehn;
#pragma unroll
    for (int k = 0; k < HID; ++k) {
      aR = fmaf(vr[k], hb[k], aR);
      aZ = fmaf(vz[k], hb[k], aZ);
      aN = fmaf(vn[k], hb[k], aN);
    }
    const float r = fsig(aR);
    const float z = fsig(aZ);
    const float nn = ftanh(fmaf(r, aN, fmaf(sn, xc2, einc)));
    hown = fmaf(z, hown - nn, nn);
#pragma unroll
    for (int k = 0; k < HID; ++k) hb[k] = rdlane(hown, k);
    // fc_out on the fresh hidden (off the recurrence critical path)
    float p = bo;
#pragma unroll
    for (int k = 0; k < HID; ++k) p = fmaf(wo[k], hb[k], p);
    if (lane < NOUT) out[t * NOUT + lane] = p;
    xc2 = xn;
  }
}

extern "C" void kernel_launch(void* const* d_in, const int* in_sizes, int n_in,
                              void* d_out, int out_size, void* d_ws, size_t ws_size,
                              hipStream_t stream) {
  (void)n_in; (void)out_size; (void)d_ws; (void)ws_size;
  const float* state = (const float*)d_in[0];
  const float* Wihe  = (const float*)d_in[1];
  const float* Whhe  = (const float*)d_in[2];
  const float* bihe  = (const float*)d_in[3];
  const float* bhhe  = (const float*)d_in[4];
  // d_in[5..7] = Wq, Wk, We: dead (softmax over length-1 sequence => alpha == 1)
  const float* Wihd  = (const float*)d_in[8];
  const float* Whhd  = (const float*)d_in[9];
  const float* bihd  = (const float*)d_in[10];
  const float* bhhd  = (const float*)d_in[11];
  const float* Wout  = (const float*)d_in[12];
  const float* bout  = (const float*)d_in[13];
  float* out = (float*)d_out;
  const int n = in_sizes[0];

  controller_gru_kernel<<<1, 32, 0, stream>>>(
      state, Wihe, Whhe, bihe, bhhe, Wihd, Whhd, bihd, bhhd, Wout, bout, out, n);
}